// ECE_18141941858507
// MI455X (gfx1250) — compile-verified
//
#include <hip/hip_runtime.h>

#define ECE_NBINS 15
#define TILE_ROWS 128      // rows per block tile (== blockDim.x, one row per lane)
#define ROW_PITCH_F 132    // 128 floats + 4 pad -> 528B row stride, 16B aligned
#define ROW_PITCH_B (ROW_PITCH_F * 4)
#define WAVES_PER_BLOCK (TILE_ROWS / 32)

#ifndef ECE_USE_ASYNC
#define ECE_USE_ASYNC 1    // gfx1250 global_load_async_to_lds_b128 staging
#endif

typedef __attribute__((ext_vector_type(2))) float v2f;
typedef __attribute__((ext_vector_type(4))) float v4f;
typedef __attribute__((ext_vector_type(8))) float v8f;
typedef __attribute__((ext_vector_type(4))) int v4i;

// ---------------------------------------------------------------------------
// Phase 0: zero the 45-float accumulator workspace (count/conf/acc per bin)
// ---------------------------------------------------------------------------
__global__ void ece_init_kernel(float* ws) {
    int t = threadIdx.x;
    if (t < 3 * ECE_NBINS) ws[t] = 0.0f;
}

// ---------------------------------------------------------------------------
// Phase 1: tile -> LDS (async), one row per lane (no cross-lane reductions),
// WMMA-based binned segment-sum of (count, confidence, accuracy).
// ---------------------------------------------------------------------------
__global__ __launch_bounds__(TILE_ROWS) void ece_main_kernel(
        const float* __restrict__ logits,
        const int* __restrict__ labels,
        float* __restrict__ ws,
        int n_rows) {
    __shared__ __align__(16) float tilebuf[TILE_ROWS * ROW_PITCH_F];
    __shared__ __align__(16) unsigned long long samplebuf[WAVES_PER_BLOCK * 32];
    __shared__ float binAcc[3 * ECE_NBINS];

    const int lrow = threadIdx.x;        // local row owned by this lane
    const int lane = threadIdx.x & 31;
    const int waveId = threadIdx.x >> 5;
    const int m = lane & 15;             // WMMA M (bin) / N (stat) index
    const int half = lane >> 4;          // 0: K=0,1   1: K=2,3
    const int numTiles = (n_rows + TILE_ROWS - 1) / TILE_ROWS;
    const int tileStride = gridDim.x;    // hoist: avoid per-tile kernarg s_load
    const int tile0 = blockIdx.x;

    // low 32 bits of a generic LDS pointer == DS byte offset (aperture layout)
    const unsigned ldsBase = (unsigned)(unsigned long long)(uintptr_t)&tilebuf[0];

    v8f acc = {};  // D[bin][stat] accumulator, striped per ISA layout

    for (int tile = tile0; tile < numTiles; tile += tileStride) {
        const int rowBase = tile * TILE_ROWS;
        const bool fullTile = (rowBase + TILE_ROWS) <= n_rows;

        // ---- stage tile: 128 rows x 512B, per-lane 16B chunks, coalesced ----
        {
            const char* gbase = (const char*)(logits + (size_t)rowBase * 128);
            if (fullTile) {
#pragma unroll
                for (int k = 0; k < 32; ++k) {
                    const int q = threadIdx.x + (k << 7);   // chunk id 0..4095
                    const unsigned loff = (unsigned)(q * 16 + ((q >> 5) << 4));
#if ECE_USE_ASYNC
                    unsigned long long ga =
                        (unsigned long long)(uintptr_t)(gbase + (size_t)q * 16);
                    unsigned la = ldsBase + loff;
                    asm volatile("global_load_async_to_lds_b128 %0, %1, off"
                                 :: "v"(la), "v"(ga) : "memory");
#else
                    v4f d = __builtin_nontemporal_load(
                        (const v4f*)(gbase + (size_t)q * 16));
                    *(v4f*)((char*)tilebuf + loff) = d;
#endif
                }
            } else {
#pragma unroll
                for (int k = 0; k < 32; ++k) {
                    const int q = threadIdx.x + (k << 7);
                    const int row = q >> 5;
                    const unsigned loff = (unsigned)(q * 16 + (row << 4));
                    if (rowBase + row < n_rows) {
#if ECE_USE_ASYNC
                        unsigned long long ga =
                            (unsigned long long)(uintptr_t)(gbase + (size_t)q * 16);
                        unsigned la = ldsBase + loff;
                        asm volatile("global_load_async_to_lds_b128 %0, %1, off"
                                     :: "v"(la), "v"(ga) : "memory");
#else
                        v4f d = __builtin_nontemporal_load(
                            (const v4f*)(gbase + (size_t)q * 16));
                        *(v4f*)((char*)tilebuf + loff) = d;
#endif
                    }
                }
            }
#if ECE_USE_ASYNC
            asm volatile("s_wait_asynccnt 0x0" ::: "memory");
#endif
        }
        __syncthreads();

        // ---- per-lane row scan: max + unstabilized sum(exp) in one pass ----
        const int grow = rowBase + lrow;
        const bool valid = grow < n_rows;
        const int lab = labels[valid ? grow : 0];

        const v4f* rowp = (const v4f*)((const char*)tilebuf + lrow * ROW_PITCH_B);
        float vmax = -3.402823466e+38f;
        float s0 = 0.0f, s1 = 0.0f;
#pragma unroll
        for (int c = 0; c < 32; ++c) {
            const v4f x = rowp[c];                     // ds_load_b128, imm off
            vmax = fmaxf(fmaxf(vmax, x.x), x.y);       // v_max3
            vmax = fmaxf(fmaxf(vmax, x.z), x.w);       // v_max3
            s0 += __expf(x.x) + __expf(x.y);
            s1 += __expf(x.z) + __expf(x.w);
        }
        const float conf = __expf(vmax) / (s0 + s1);   // = max softmax prob

        // accuracy: pred==label  <=>  logits[row][label]==rowmax (ties ~ never)
        const float xl =
            *(const float*)((const char*)tilebuf + lrow * ROW_PITCH_B + lab * 4);

        int b = (int)ceilf(conf * (float)ECE_NBINS) - 1;
        b = min(max(b, 0), ECE_NBINS - 1);
        // sanitize invalid rows BEFORE WMMA (NaN would poison D even via 0*NaN)
        const float confS = valid ? conf : 0.0f;
        const int hitBit = (valid && xl == vmax) ? 1 : 0;
        // key: bin[3:0] | hit<<4 ; invalid rows -> bin 15 ("trash" row of D,
        // never read at extraction), conf forced to 0 so no NaN enters B.
        const int key = valid ? (b | (hitBit << 4)) : 15;

        // ---- publish per-wave samples: {key, conf} 8B records in LDS ----
        samplebuf[waveId * 32 + lane] =
            ((unsigned long long)__float_as_uint(confS) << 32) | (unsigned)key;

        // ---- WMMA binned scatter: 8 x (4 samples) per wave-32 samples ----
        // A (16x4): A[m][k] = (bin_k == m); B (4x16): B[k][n] = stat_n(k)
        const char* sb = (const char*)&samplebuf[waveId * 32];
#pragma unroll
        for (int g = 0; g < 8; ++g) {
            // samples j0 = 4g+2*half, j1 = j0+1 are adjacent 8B records:
            // one broadcast ds_load_b128 -> {key0, conf0, key1, conf1}
            const v4i p = *(const v4i*)(sb + g * 32 + half * 16);
            const int k0 = p.x, k1 = p.z;
            const float c0 = __int_as_float(p.y);
            const float c1 = __int_as_float(p.w);
            v2f A, B;
            A.x = ((k0 & 15) == m) ? 1.0f : 0.0f;
            A.y = ((k1 & 15) == m) ? 1.0f : 0.0f;
            const float h0 = (k0 & 16) ? 1.0f : 0.0f;
            const float h1 = (k1 & 16) ? 1.0f : 0.0f;
            B.x = (m == 0) ? 1.0f : (m == 1) ? c0 : (m == 2) ? h0 : 0.0f;
            B.y = (m == 0) ? 1.0f : (m == 1) ? c1 : (m == 2) ? h1 : 0.0f;
            acc = __builtin_amdgcn_wmma_f32_16x16x4_f32(
                false, A, false, B, (short)0, acc, false, false);
        }
        __syncthreads();  // tile buffer free for next stage
    }

    // ---- fold per-wave D into per-block LDS, then 45 global atomics/block ----
    for (int t = threadIdx.x; t < 3 * ECE_NBINS; t += blockDim.x) binAcc[t] = 0.0f;
    __syncthreads();
    // C/D layout: lane n<16 holds D[0..7][n] in VGPR0..7; lane 16+n -> D[8..15][n]
    if (m < 3) {
        const int stat = m;
        const int baseBin = half * 8;
#pragma unroll
        for (int v = 0; v < 8; ++v) {
            const int bb = baseBin + v;
            if (bb < ECE_NBINS) atomicAdd(&binAcc[bb * 3 + stat], acc[v]);
        }
    }
    __syncthreads();
    for (int t = threadIdx.x; t < 3 * ECE_NBINS; t += blockDim.x) {
        const float p = binAcc[t];
        if (p != 0.0f) atomicAdd(&ws[t], p);
    }
}

// ---------------------------------------------------------------------------
// Phase 2: 15-bin ECE scalar (all rows covered by phase 1 incl. tail)
// ---------------------------------------------------------------------------
__global__ void ece_final_kernel(const float* __restrict__ ws,
                                 float* __restrict__ out, int n_rows) {
    if (threadIdx.x == 0 && blockIdx.x == 0) {
        float ece = 0.0f;
        const float n = (float)n_rows;
        for (int b = 0; b < ECE_NBINS; ++b) {
            const float cnt = ws[b * 3 + 0];
            const float sc  = ws[b * 3 + 1];
            const float sa  = ws[b * 3 + 2];
            if (cnt > 0.0f) ece += fabsf((sc - sa) / cnt) * (cnt / n);
        }
        out[0] = ece;
    }
}

extern "C" void kernel_launch(void* const* d_in, const int* in_sizes, int n_in,
                              void* d_out, int out_size, void* d_ws,
                              size_t ws_size, hipStream_t stream) {
    const float* logits = (const float*)d_in[0];
    const int* labels = (const int*)d_in[1];
    float* out = (float*)d_out;
    float* ws = (float*)d_ws;
    const int n_rows = in_sizes[1];  // 1,000,000 ; C fixed at 128

    ece_init_kernel<<<1, 64, 0, stream>>>(ws);
    ece_main_kernel<<<2048, TILE_ROWS, 0, stream>>>(logits, labels, ws, n_rows);
    ece_final_kernel<<<1, 32, 0, stream>>>(ws, out, n_rows);
}